// BatchGRU_1245540515928
// MI455X (gfx1250) — compile-verified
//
#include <hip/hip_runtime.h>

typedef __attribute__((ext_vector_type(2))) float v2f;
typedef __attribute__((ext_vector_type(8))) float v8f;

#define H       300
#define HP      304           // H padded to multiple of 16
#define LMAX    64
#define NGRAPH  1024
#define KP      608           // fused K: [x (304) | h (304)]
#define NG      1216          // gate cols: 4 blocks of 304 (r, z, n_x, n_h)
#define NT      76            // NG / 16
#define KT      152           // KP / 4 (WMMA K-steps)
#define MROWS   32            // graphs per workgroup
#define XS      610           // xh LDS row stride (floats), padded vs 608 for banks
#define WSZ_ELEMS (NT * KT * 64)   // 739328 floats per direction (swizzled Wcat)

// ---------------------------------------------------------------------------
// Kernel 0: build swizzled weight fragments + fused bias, one direction.
// Swizzled layout: float2 index ((nt*KT + k)*32 + lane), matching the
// per-lane B-fragment layout of V_WMMA_F32_16X16X4_F32, so the GRU hot loop
// streams weights with fully coalesced global_load_b64.
// Gate blocks (n = blk*304 + j): 0=r(fused ih+hh), 1=z(fused), 2=n_x, 3=n_h.
// K: koff<300 -> x-side (w_ih), 304<=koff<604 -> h-side (w_hh), else 0.
// ---------------------------------------------------------------------------
__global__ void prep_weights(float* __restrict__ wsz, float* __restrict__ bsum,
                             const float* __restrict__ w_ih,
                             const float* __restrict__ w_hh,
                             const float* __restrict__ b_ih,
                             const float* __restrict__ b_hh) {
  int s = blockIdx.x * blockDim.x + threadIdx.x;
  if (s >= WSZ_ELEMS) return;
  int e    = s & 1;
  int lane = (s >> 1) & 31;
  int t3   = s >> 6;
  int k    = t3 % KT;
  int nt   = t3 / KT;
  int n    = nt * 16 + (lane & 15);
  int koff = 4 * k + 2 * (lane >> 4) + e;

  int blk = n / HP;
  int j   = n - blk * HP;
  float v = 0.f;
  if (j < H) {
    int g = (blk == 1) ? (H + j) : ((blk >= 2) ? (2 * H + j) : j);
    if (koff < H) {                            // x-side
      if (blk != 3) v = w_ih[g * H + koff];
    } else if (koff >= HP && koff < HP + H) {  // h-side
      if (blk != 2) v = w_hh[g * H + (koff - HP)];
    }
  }
  wsz[s] = v;

  if (s < NG) {
    int blk2 = s / HP, j2 = s - blk2 * HP;
    float bv = 0.f;
    if (j2 < H) {
      if (blk2 == 0)      bv = b_ih[j2] + b_hh[j2];
      else if (blk2 == 1) bv = b_ih[H + j2] + b_hh[H + j2];
      else if (blk2 == 2) bv = b_ih[2 * H + j2];
      else                bv = b_hh[2 * H + j2];
    }
    bsum[s] = bv;
  }
}

// ---------------------------------------------------------------------------
// Segment extent recovery from (batch, pos).
// ---------------------------------------------------------------------------
__global__ void seg_init(int* __restrict__ seg_start, int* __restrict__ seg_len) {
  int i = blockIdx.x * blockDim.x + threadIdx.x;
  if (i < NGRAPH) { seg_start[i] = 0; seg_len[i] = 0; }
}

__global__ void seg_scan(const int* __restrict__ batch, const int* __restrict__ pos,
                         int n, int* __restrict__ seg_start, int* __restrict__ seg_len) {
  int i = blockIdx.x * blockDim.x + threadIdx.x;
  if (i >= n) return;
  int b = batch[i], p = pos[i];
  atomicMax(&seg_len[b], p + 1);
  if (p == 0) seg_start[b] = i;   // benign multi-write of identical value never occurs (unique pos 0)
}

// ---------------------------------------------------------------------------
// Kernel 2: dense padded message buffer [B][LMAX][HP] = relu(node+bias) and
// segment-max h0 [B][HP] (zero in pad columns). One block per graph.
// ---------------------------------------------------------------------------
__global__ void pad_h0(const float* __restrict__ node, const float* __restrict__ bias,
                       const int* __restrict__ seg_start, const int* __restrict__ seg_len,
                       float* __restrict__ padded, float* __restrict__ h0) {
  const int b = blockIdx.x;
  const int start = seg_start[b];
  const int len   = seg_len[b];
  for (int idx = threadIdx.x; idx < LMAX * HP; idx += blockDim.x) {
    int t = idx / HP, c = idx - t * HP;
    float v = 0.f;
    if (t < len && c < H) {
      float x = node[(size_t)(start + t) * H + c] + bias[c];
      v = x > 0.f ? x : 0.f;
    }
    padded[(size_t)b * LMAX * HP + idx] = v;
  }
  for (int c = threadIdx.x; c < HP; c += blockDim.x) {
    float m = 0.f;
    if (c < H && len > 0) {
      m = -3.402823466e38f;
      for (int t = 0; t < len; ++t)
        m = fmaxf(m, node[(size_t)(start + t) * H + c]);
    }
    h0[(size_t)b * HP + c] = m;
  }
}

// ---------------------------------------------------------------------------
// Kernel 3: fused recurrent GRU. One workgroup = (direction, 32-graph tile),
// iterates all 64 time steps with h resident in LDS. Per step:
//   gates[32,1216] = [x_t | h][32,608] @ Wcat^T  (+ fused bias)  via
//   v_wmma_f32_16x16x4_f32, B-fragment reused across two M-tiles.
// ---------------------------------------------------------------------------
__global__ __launch_bounds__(256, 1) void gru_kernel(
    const float* __restrict__ wsz_f, const float* __restrict__ wsz_b,
    const float* __restrict__ bsum_f, const float* __restrict__ bsum_b,
    const float* __restrict__ padded, const float* __restrict__ h0,
    const int* __restrict__ seg_start, const int* __restrict__ seg_len,
    float* __restrict__ out) {
  extern __shared__ float smem[];
  float* xh    = smem;                 // [MROWS][XS]   cols [0,304)=x_t, [304,608)=h
  float* gates = smem + MROWS * XS;    // [MROWS][NG]

  const int dir  = blockIdx.x & 1;
  const int rb   = blockIdx.x >> 1;
  const int row0 = rb * MROWS;
  const float* __restrict__ wsz  = dir ? wsz_b  : wsz_f;
  const float* __restrict__ bsum = dir ? bsum_b : bsum_f;
  const v2f*   __restrict__ W2   = (const v2f*)wsz;

  const int tid  = threadIdx.x;
  const int wave = tid >> 5;
  const int lane = tid & 31;
  const int ml   = lane & 15;
  const int kh   = (lane >> 4) << 1;   // 0 for lanes 0-15, 2 for lanes 16-31
  const int mrow = (lane >> 4) * 8;    // C/D fragment row base

  // init hidden state from h0 (HP cols incl. zero pads)
  for (int idx = tid; idx < MROWS * HP; idx += 256) {
    int m = idx / HP, c = idx - m * HP;
    xh[m * XS + HP + c] = h0[(size_t)(row0 + m) * HP + c];
  }

  for (int step = 0; step < LMAX; ++step) {
    const int t = dir ? (LMAX - 1 - step) : step;
    __syncthreads();  // prior GEMM/elementwise done before x-cols are overwritten
    // stage x_t into LDS
    for (int idx = tid; idx < MROWS * HP; idx += 256) {
      int m = idx / HP, c = idx - m * HP;
      xh[m * XS + c] = padded[((size_t)(row0 + m) * LMAX + t) * HP + c];
    }
    __syncthreads();

    // ---- WMMA GEMM: 8 waves sweep the 76 N-tiles ----
    for (int nt = wave; nt < NT; nt += 8) {
      const int n = nt * 16 + ml;
      const float bb = bsum[n];
      v8f acc0 = {bb, bb, bb, bb, bb, bb, bb, bb};  // bias is per-column
      v8f acc1 = acc0;
      const v2f* wp = W2 + (size_t)nt * KT * 32 + lane;
      const float* a0p = &xh[ml * XS + kh];
      const float* a1p = &xh[(ml + 16) * XS + kh];
#pragma unroll 4
      for (int k = 0; k < KT; ++k) {
        v2f bfrag = wp[k * 32];                       // coalesced global_load_b64 (L2)
        v2f a0 = *(const v2f*)(a0p + 4 * k);          // ds_load_b64
        v2f a1 = *(const v2f*)(a1p + 4 * k);
        acc0 = __builtin_amdgcn_wmma_f32_16x16x4_f32(
            false, a0, false, bfrag, (short)0, acc0, false, false);
        acc1 = __builtin_amdgcn_wmma_f32_16x16x4_f32(
            false, a1, false, bfrag, (short)0, acc1, false, false);
      }
#pragma unroll
      for (int r = 0; r < 8; ++r) {
        gates[(mrow + r) * NG + n]        = acc0[r];
        gates[(16 + mrow + r) * NG + n]   = acc1[r];
      }
    }
    __syncthreads();

    // ---- gate nonlinearity, h update, scatter to final output ----
    for (int idx = tid; idx < MROWS * H; idx += 256) {
      int m = idx / H, c = idx - m * H;
      float gr  = gates[m * NG + c];            // reset  (x+h fused)
      float gz  = gates[m * NG + HP + c];       // update (x+h fused)
      float gnx = gates[m * NG + 2 * HP + c];   // new, x-side
      float gnh = gates[m * NG + 3 * HP + c];   // new, h-side
      float r  = 1.f / (1.f + __expf(-gr));
      float z  = 1.f / (1.f + __expf(-gz));
      float nn = tanhf(gnx + r * gnh);
      float hp = xh[m * XS + HP + c];
      float hn = (1.f - z) * nn + z * hp;
      xh[m * XS + HP + c] = hn;
      int b = row0 + m;
      if (t < seg_len[b])
        out[(size_t)(seg_start[b] + t) * (2 * H) + dir * H + c] = hn;
    }
  }
}

// ---------------------------------------------------------------------------
extern "C" void kernel_launch(void* const* d_in, const int* in_sizes, int n_in,
                              void* d_out, int out_size, void* d_ws, size_t ws_size,
                              hipStream_t stream) {
  (void)n_in; (void)out_size; (void)ws_size;
  const float* node   = (const float*)d_in[0];
  const int*   batch  = (const int*)d_in[1];
  const int*   pos    = (const int*)d_in[2];
  const float* bias   = (const float*)d_in[3];
  const float* w_ih_f = (const float*)d_in[4];
  const float* w_hh_f = (const float*)d_in[5];
  const float* b_ih_f = (const float*)d_in[6];
  const float* b_hh_f = (const float*)d_in[7];
  const float* w_ih_b = (const float*)d_in[8];
  const float* w_hh_b = (const float*)d_in[9];
  const float* b_ih_b = (const float*)d_in[10];
  const float* b_hh_b = (const float*)d_in[11];
  float* out = (float*)d_out;
  const int N = in_sizes[0] / H;

  // workspace carve-up (256B aligned)
  char* ws = (char*)d_ws;
  size_t off = 0;
  auto carve = [&](size_t bytes) -> void* {
    void* p = ws + off;
    off = (off + bytes + 255) & ~(size_t)255;
    return p;
  };
  float* wsz_f  = (float*)carve(sizeof(float) * WSZ_ELEMS);
  float* wsz_b  = (float*)carve(sizeof(float) * WSZ_ELEMS);
  float* bsum_f = (float*)carve(sizeof(float) * NG);
  float* bsum_b = (float*)carve(sizeof(float) * NG);
  int*   segs   = (int*)carve(sizeof(int) * NGRAPH);
  int*   segl   = (int*)carve(sizeof(int) * NGRAPH);
  float* h0     = (float*)carve(sizeof(float) * (size_t)NGRAPH * HP);
  float* padded = (float*)carve(sizeof(float) * (size_t)NGRAPH * LMAX * HP);

  const int pgrid = (WSZ_ELEMS + 255) / 256;
  prep_weights<<<pgrid, 256, 0, stream>>>(wsz_f, bsum_f, w_ih_f, w_hh_f, b_ih_f, b_hh_f);
  prep_weights<<<pgrid, 256, 0, stream>>>(wsz_b, bsum_b, w_ih_b, w_hh_b, b_ih_b, b_hh_b);
  seg_init<<<(NGRAPH + 255) / 256, 256, 0, stream>>>(segs, segl);
  seg_scan<<<(N + 255) / 256, 256, 0, stream>>>(batch, pos, N, segs, segl);
  pad_h0<<<NGRAPH, 256, 0, stream>>>(node, bias, segs, segl, padded, h0);

  const size_t smem_bytes = (size_t)(MROWS * XS + MROWS * NG) * sizeof(float); // ~234 KB
  gru_kernel<<<(NGRAPH / MROWS) * 2, 256, smem_bytes, stream>>>(
      wsz_f, wsz_b, bsum_f, bsum_b, padded, h0, segs, segl, out);
}